// SplitAndMeanPooling_2911987826810
// MI455X (gfx1250) — compile-verified
//
#include <hip/hip_runtime.h>

typedef __attribute__((ext_vector_type(2))) float v2f;
typedef __attribute__((ext_vector_type(8))) float v8f;

constexpr int D = 128;   // feature dim fixed by the reference; enables immediate-offset loads

__device__ __forceinline__ v8f wmma_acc(v2f a, v2f b, v8f c) {
    // D = A(16x4, all-ones) * B(4x16) + C  -> every row of D = per-column sums of B
    return __builtin_amdgcn_wmma_f32_16x16x4_f32(false, a, false, b, (short)0, c, false, false);
}

// ---------------------------------------------------------------------------
// Kernel 1: exclusive prefix-sum of per-graph sizes -> row offsets (d_ws).
// ---------------------------------------------------------------------------
__global__ void __launch_bounds__(256) scan_sizes_kernel(const int* __restrict__ sizes,
                                                         int* __restrict__ offs, int G) {
    __shared__ int partial[256];
    const int t = threadIdx.x;
    const int chunk = (G + 255) / 256;
    const int start = t * chunk;
    const int end   = (start + chunk < G) ? (start + chunk) : G;

    int s = 0;
    for (int i = start; i < end; ++i) s += sizes[i];
    partial[t] = s;
    __syncthreads();

    if (t == 0) {
        int acc = 0;
        for (int i = 0; i < 256; ++i) { int v = partial[i]; partial[i] = acc; acc += v; }
    }
    __syncthreads();

    int acc = partial[t];
    for (int i = start; i < end; ++i) { offs[i] = acc; acc += sizes[i]; }
    if (t == 255) offs[G] = acc;
}

// ---------------------------------------------------------------------------
// Kernel 2: segment-mean via V_WMMA_F32_16X16X4_F32, all-ones A.
// One block (8 wave32 waves) per graph; wave w reduces columns [16w,16w+16).
// 16 rows per main iteration: 8 global_load_b32 with immediate offsets off a
// single base pointer, then 4 chained WMMAs -> staggered loadcnt waits.
// ---------------------------------------------------------------------------
__global__ void __launch_bounds__(256) seg_mean_wmma_kernel(const float* __restrict__ feat,
                                                            const int* __restrict__ offs,
                                                            const int* __restrict__ sizes,
                                                            float* __restrict__ out) {
    const int g    = blockIdx.x;
    const int lane = threadIdx.x & 31;
    const int wave = threadIdx.x >> 5;           // 0..7 -> column slice
    const int col  = (wave << 4) + (lane & 15);  // feature column this lane feeds
    const int half = lane >> 4;                  // 0: rows +0/+1, 1: rows +2/+3

    const int row0 = offs[g];
    const int n    = sizes[g];                   // uniform across the block

    // lane's streaming pointer: row (row0 + half*2), this lane's column
    const float* q = feat + ((size_t)(row0 + half * 2) * D + (size_t)col);

    v2f a; a[0] = 1.0f; a[1] = 1.0f;
    v8f c = {0.f, 0.f, 0.f, 0.f, 0.f, 0.f, 0.f, 0.f};

    const int nfull = n & ~3;
    int k = 0;

    // main loop: 16 rows / iteration (all offsets are compile-time immediates)
    for (; k + 16 <= nfull; k += 16, q += 16 * D) {
        if (k + 32 <= nfull) {                   // uniform guard; EXEC stays all-1s
            __builtin_prefetch(q + 16 * D, 0, 0);   // global_prefetch_b8 (next tile)
            __builtin_prefetch(q + 24 * D, 0, 0);
        }
        v2f b0, b1, b2, b3;
        b0[0] = q[ 0 * D];  b0[1] = q[ 1 * D];
        b1[0] = q[ 4 * D];  b1[1] = q[ 5 * D];
        b2[0] = q[ 8 * D];  b2[1] = q[ 9 * D];
        b3[0] = q[12 * D];  b3[1] = q[13 * D];
        c = wmma_acc(a, b0, c);
        c = wmma_acc(a, b1, c);
        c = wmma_acc(a, b2, c);
        c = wmma_acc(a, b3, c);
    }
    // remainder groups of 4 rows
    for (; k < nfull; k += 4, q += 4 * D) {
        v2f b;
        b[0] = q[0];
        b[1] = q[D];
        c = wmma_acc(a, b, c);
    }
    // tail (n % 4 rows), zero-padded; n is block-uniform so branch is uniform
    if (k < n) {
        const int r0 = k + half * 2;
        v2f b;
        b[0] = (r0     < n) ? q[0] : 0.0f;
        b[1] = (r0 + 1 < n) ? q[D] : 0.0f;
        c = wmma_acc(a, b, c);
    }

    const float inv = (n > 0) ? (1.0f / (float)n) : 0.0f;
    // C/D layout: accumulator VGPR0, lanes 0-15 hold D[M=0, N=lane] = column sums
    if (lane < 16) out[(size_t)g * D + (size_t)col] = c[0] * inv;
}

// ---------------------------------------------------------------------------
// Kernel 3: pass-throughs. lapl as f32; sizes as float values (JAX promotion).
// ---------------------------------------------------------------------------
__global__ void tail_copy_kernel(const float* __restrict__ lapl,
                                 const int* __restrict__ sizes,
                                 float* __restrict__ out, int G) {
    const int i = blockIdx.x * blockDim.x + threadIdx.x;
    const size_t baseMean = (size_t)G * (size_t)D;
    if (i < 4)  out[baseMean + (size_t)i] = lapl[i];
    if (i < G)  out[baseMean + 4 + (size_t)i] = (float)sizes[i];
}

// ---------------------------------------------------------------------------
extern "C" void kernel_launch(void* const* d_in, const int* in_sizes, int n_in,
                              void* d_out, int out_size, void* d_ws, size_t ws_size,
                              hipStream_t stream) {
    const float* feat  = (const float*)d_in[0];  // [N, 128] f32
    const float* lapl  = (const float*)d_in[1];  // [4] f32 pass-through
    const int*   sizes = (const int*)d_in[2];    // [G] i32
    float*       out   = (float*)d_out;
    int*         offs  = (int*)d_ws;             // (G+1) ints of scratch

    const int G = in_sizes[2];

    scan_sizes_kernel<<<1, 256, 0, stream>>>(sizes, offs, G);
    seg_mean_wmma_kernel<<<G, 256, 0, stream>>>(feat, offs, sizes, out);

    const int tailN = (G > 4) ? G : 4;
    tail_copy_kernel<<<(tailN + 255) / 256, 256, 0, stream>>>(lapl, sizes, out, G);
}